// IDEGM_39135742001540
// MI455X (gfx1250) — compile-verified
//
#include <hip/hip_runtime.h>
#include <math.h>
#include <stdint.h>

// ---------------------------------------------------------------------------
// Problem constants (B,C,H,W) = (4,128,96,96)
// ---------------------------------------------------------------------------
#define CCH 128
#define BB  4
#define HH  96
#define WW  96
#define HW  (HH*WW)                       // 9216
#define NPIX ((size_t)BB*CCH*HW)          // 4718592 per tensor
#define AI  97                            // padded axial length (i axis)
#define AJ  384                           // sequence axis (j / k)
#define AROWS ((size_t)BB*AI*AJ)          // 148992 LN rows
#define TSZ (AROWS*(size_t)CCH)           // 19070976 attn-layout tensor
#define EPSV 1e-5f

typedef float v2f __attribute__((ext_vector_type(2)));
typedef float v8f __attribute__((ext_vector_type(8)));

// CDNA5 fp32 WMMA: D(16x16,f32) = A(16x4,f32) * B(4x16,f32) + C
static __device__ __forceinline__ v8f wmma4(v2f a, v2f b, v8f c) {
  return __builtin_amdgcn_wmma_f32_16x16x4_f32(false, a, false, b,
                                               (short)0, c, false, false);
}

static __device__ __forceinline__ float gelu_exact(float v) {
  return 0.5f * v * (1.0f + erff(v * 0.70710678118654752f));
}

// CDNA5 async global->LDS copy (ASYNCcnt-tracked), per CDNA5 ISA.
// LDS offset goes in the vdst VGPR, the 64-bit global address in vaddr
static __device__ __forceinline__ void async_ld_f32_to_lds(uint32_t lds_off,
                                                           const float* g) {
  asm volatile("global_load_async_to_lds_b32 %0, %1, off"
               :: "v"(lds_off), "v"((uint64_t)(uintptr_t)g)
               : "memory");
}
static __device__ __forceinline__ void wait_async0() {
  asm volatile("s_wait_asynccnt 0x0" ::: "memory");
}

// ---------------------------------------------------------------------------
// 1x1 conv (channel GEMM) + BN + GELU (+ optional 2nd input, + residual)
// Out[b,o,n] = gelu(bn(sum_c W[o,c]*(X1[b,c,n]+X2[b,c,n]))) + resid[b,o,n]
// grid (HW/256, CCH/16, BB), block 128; wave owns a 16x64 output strip
// (one weight-tile load feeds 4 WMMAs). BN params staged to LDS via async
// global->LDS copy overlapped with the GEMM loop.
// ---------------------------------------------------------------------------
template <bool HAS2, bool HASR>
__global__ void k_conv_bn_gelu_t(const float* __restrict__ X1,
                                 const float* __restrict__ X2,
                                 const float* __restrict__ Wm,
                                 const float* __restrict__ gain,
                                 const float* __restrict__ beta,
                                 const float* __restrict__ rmean,
                                 const float* __restrict__ rvar,
                                 const float* __restrict__ resid,
                                 float* __restrict__ Out) {
  __shared__ float sbn[64];   // [0:16) gain, [16:32) beta, [32:48) rm, [48:64) rv
  const int lane  = threadIdx.x & 31;
  const int wv    = threadIdx.x >> 5;
  const int b     = blockIdx.z;
  const int o0    = blockIdx.y * 16;
  const int nbase = blockIdx.x * 256 + wv * 64;
  const size_t base = (size_t)b * CCH * HW;
  const int m  = lane & 15;          // A row (M) / B col (N) within tile
  const int kb = (lane >> 4) * 2;    // K pair selected by lane[4]

  if (threadIdx.x < 64) {            // waves 0,1 entirely: EXEC uniform
    int t = threadIdx.x;
    const float* p;
    switch (t >> 4) {
      case 0:  p = gain;  break;
      case 1:  p = beta;  break;
      case 2:  p = rmean; break;
      default: p = rvar;  break;
    }
    async_ld_f32_to_lds((uint32_t)(uintptr_t)(&sbn[t]), p + o0 + (t & 15));
  }

  const v8f z = {0.f,0.f,0.f,0.f,0.f,0.f,0.f,0.f};
  v8f acc[4] = {z, z, z, z};
  for (int c0 = 0; c0 < CCH; c0 += 4) {
    const float* wp = Wm + (size_t)(o0 + m) * CCH + c0 + kb;
    v2f a; a.x = wp[0]; a.y = wp[1];
    const size_t xrow = base + (size_t)(c0 + kb) * HW;
#pragma unroll
    for (int t = 0; t < 4; ++t) {
      size_t xi = xrow + (size_t)(nbase + t * 16 + m);
      v2f bv;
      bv.x = X1[xi]; bv.y = X1[xi + HW];
      if (HAS2) { bv.x += X2[xi]; bv.y += X2[xi + HW]; }
      acc[t] = wmma4(a, bv, acc[t]);
    }
  }

  wait_async0();
  __syncthreads();

  const int mo = (lane >> 4) * 8;
#pragma unroll
  for (int t = 0; t < 4; ++t) {
    const int n = nbase + t * 16 + m;
#pragma unroll
    for (int r = 0; r < 8; ++r) {
      int oc = mo + r;                       // channel within tile (0..15)
      int o  = o0 + oc;
      float inv = sbn[oc] * rsqrtf(sbn[48 + oc] + EPSV);
      float v = (acc[t][r] - sbn[32 + oc]) * inv + sbn[16 + oc];
      v = gelu_exact(v);
      size_t oi = base + (size_t)o * HW + n;
      if (HASR) v += resid[oi];
      Out[oi] = v;
    }
  }
}

// ---------------------------------------------------------------------------
// Pack h-path: T[b,i,j,c] (i<97, j<384) from xh/yh [b,c,96,96]
// j quadrants: 0:xh(l1) 1:yh(l1) 2:xh(l2) 3:yh(l2); l1 row=i-1, l2 row=i
// ---------------------------------------------------------------------------
__global__ void k_pack_h(const float* __restrict__ xh, const float* __restrict__ yh,
                         float* __restrict__ T) {
  size_t idx = (size_t)blockIdx.x * blockDim.x + threadIdx.x;
  if (idx >= TSZ) return;
  int c = (int)(idx & 127);
  int j = (int)((idx >> 7) % AJ);
  int i = (int)(((idx >> 7) / AJ) % AI);
  int b = (int)(idx / ((size_t)128 * AJ * AI));
  int q = j / 96, w = j % 96;
  const float* src = (q & 1) ? yh : xh;
  int h = (q < 2) ? (i - 1) : i;
  float v = 0.f;
  if (h >= 0 && h < HH)
    v = src[(((size_t)b * CCH + c) * HH + h) * WW + w];
  T[idx] = v;
}

// Pack v-path: T[b,i,r,c] (i<97, r<384) from xv/yv
__global__ void k_pack_v(const float* __restrict__ xv, const float* __restrict__ yv,
                         float* __restrict__ T) {
  size_t idx = (size_t)blockIdx.x * blockDim.x + threadIdx.x;
  if (idx >= TSZ) return;
  int c = (int)(idx & 127);
  int r = (int)((idx >> 7) % AJ);
  int i = (int)(((idx >> 7) / AJ) % AI);
  int b = (int)(idx / ((size_t)128 * AJ * AI));
  int wcol, row;
  if (r < 192) { wcol = i - 1; row = r; } else { wcol = i; row = r - 192; }
  float v = 0.f;
  if (wcol >= 0 && wcol < WW) {
    const float* src = (row < HH) ? xv : yv;
    int hh = (row < HH) ? row : row - HH;
    v = src[(((size_t)b * CCH + c) * HH + hh) * WW + wcol];
  }
  T[idx] = v;
}

// ---------------------------------------------------------------------------
// LayerNorm over last dim (128), in place. Wave per row, float4 per lane.
// ---------------------------------------------------------------------------
__global__ void k_layernorm(float* __restrict__ T, const float* __restrict__ lng,
                            const float* __restrict__ lnb) {
  int row  = blockIdx.x * (blockDim.x >> 5) + (threadIdx.x >> 5);
  int lane = threadIdx.x & 31;
  float4 v = ((const float4*)(T + (size_t)row * 128))[lane];
  float s  = v.x + v.y + v.z + v.w;
  float sq = v.x*v.x + v.y*v.y + v.z*v.z + v.w*v.w;
#pragma unroll
  for (int off = 16; off; off >>= 1) {
    s  += __shfl_xor(s,  off, 32);
    sq += __shfl_xor(sq, off, 32);
  }
  float mu  = s * (1.f / 128.f);
  float var = sq * (1.f / 128.f) - mu * mu;
  float inv = rsqrtf(var + EPSV);
  float4 g  = ((const float4*)lng)[lane];
  float4 bb = ((const float4*)lnb)[lane];
  v.x = (v.x - mu) * inv * g.x + bb.x;
  v.y = (v.y - mu) * inv * g.y + bb.y;
  v.z = (v.z - mu) * inv * g.z + bb.z;
  v.w = (v.w - mu) * inv * g.w + bb.w;
  ((float4*)(T + (size_t)row * 128))[lane] = v;
}

// ---------------------------------------------------------------------------
// QKV: Q[n,o] = sum_c XN[n,c]*Wq[o,c] (and K,V). One A tile feeds 3 WMMAs.
// grid (AROWS/16, 2), block 128; wave handles o-tile (blockIdx.y*4+wv)
// ---------------------------------------------------------------------------
__global__ void k_qkv(const float* __restrict__ XN,
                      const float* __restrict__ Wq, const float* __restrict__ Wk,
                      const float* __restrict__ Wv,
                      float* __restrict__ Q, float* __restrict__ K,
                      float* __restrict__ V) {
  const int lane = threadIdx.x & 31;
  const int wv   = threadIdx.x >> 5;
  const int r0   = blockIdx.x * 16;
  const int o0   = (blockIdx.y * 4 + wv) * 16;
  const int m  = lane & 15;
  const int kb = (lane >> 4) * 2;
  const v8f z = {0.f,0.f,0.f,0.f,0.f,0.f,0.f,0.f};
  v8f aq = z, ak = z, av = z;
  for (int c0 = 0; c0 < 128; c0 += 4) {
    v2f a  = *(const v2f*)(XN + (size_t)(r0 + m) * 128 + c0 + kb);
    v2f bq = *(const v2f*)(Wq + (size_t)(o0 + m) * 128 + c0 + kb);
    v2f bk = *(const v2f*)(Wk + (size_t)(o0 + m) * 128 + c0 + kb);
    v2f bv = *(const v2f*)(Wv + (size_t)(o0 + m) * 128 + c0 + kb);
    aq = wmma4(a, bq, aq);
    ak = wmma4(a, bk, ak);
    av = wmma4(a, bv, av);
  }
  const int mo = (lane >> 4) * 8;
#pragma unroll
  for (int r = 0; r < 8; ++r) {
    size_t oi = (size_t)(r0 + mo + r) * 128 + o0 + m;
    Q[oi] = aq[r]; K[oi] = ak[r]; V[oi] = av[r];
  }
}

// ---------------------------------------------------------------------------
// Scores: per (b,i): S[j,k] = (1/(97*128)^2) * sum_c Q[j,c]*K[k,c]
// grid (AJ/64, AJ/64, BB*AI), block 128; wave: 16(j) x 64(k), Q tile reused x4
// ---------------------------------------------------------------------------
__global__ void k_scores(const float* __restrict__ Q, const float* __restrict__ K,
                         float* __restrict__ S) {
  const int lane  = threadIdx.x & 31;
  const int wv    = threadIdx.x >> 5;
  const int bi    = blockIdx.z;
  const int j0    = (blockIdx.x * 4 + wv) * 16;
  const int kbase = blockIdx.y * 64;
  const size_t base = (size_t)bi * AJ * 128;
  const int m  = lane & 15;
  const int kb = (lane >> 4) * 2;
  const v8f z = {0.f,0.f,0.f,0.f,0.f,0.f,0.f,0.f};
  v8f acc[4] = {z, z, z, z};
  for (int c0 = 0; c0 < 128; c0 += 4) {
    v2f a = *(const v2f*)(Q + base + (size_t)(j0 + m) * 128 + c0 + kb);
#pragma unroll
    for (int t = 0; t < 4; ++t) {
      v2f b = *(const v2f*)(K + base + (size_t)(kbase + t * 16 + m) * 128 + c0 + kb);
      acc[t] = wmma4(a, b, acc[t]);
    }
  }
  const float invs = 1.f / (12416.f * 12416.f);   // (97*128)^2
  const size_t sb = (size_t)bi * AJ * AJ;
  const int mo = (lane >> 4) * 8;
#pragma unroll
  for (int t = 0; t < 4; ++t)
#pragma unroll
    for (int r = 0; r < 8; ++r)
      S[sb + (size_t)(j0 + mo + r) * AJ + kbase + t * 16 + m] = acc[t][r] * invs;
}

// ---------------------------------------------------------------------------
// Softmax over i (stride AJ*AJ), per (b,j,k). Coalesced across k.
// ---------------------------------------------------------------------------
__global__ void k_softmax_i(float* __restrict__ S) {
  const size_t JK = (size_t)AJ * AJ;
  size_t t = (size_t)blockIdx.x * blockDim.x + threadIdx.x;
  if (t >= (size_t)BB * JK) return;
  int b = (int)(t / JK);
  size_t jk = t % JK;
  float* p = S + (size_t)b * AI * JK + jk;
  float mx = -INFINITY;
  for (int i = 0; i < AI; ++i) mx = fmaxf(mx, p[(size_t)i * JK]);
  float sum = 0.f;
  for (int i = 0; i < AI; ++i) sum += __expf(p[(size_t)i * JK] - mx);
  float inv = 1.f / sum;
  for (int i = 0; i < AI; ++i) {
    float e = __expf(p[(size_t)i * JK] - mx);
    p[(size_t)i * JK] = e * inv;
  }
}

// ---------------------------------------------------------------------------
// AV: per (b,i): O[j,c] = sum_k P[j,k]*V[k,c]
// grid (AJ/64, BB*AI), block 128; wave: 16(j) x 128(c), P tile reused x8
// ---------------------------------------------------------------------------
__global__ void k_av(const float* __restrict__ P, const float* __restrict__ V,
                     float* __restrict__ O) {
  const int lane = threadIdx.x & 31;
  const int wv   = threadIdx.x >> 5;
  const int bi   = blockIdx.y;
  const int j0   = (blockIdx.x * 4 + wv) * 16;
  const size_t pb = (size_t)bi * AJ * AJ;
  const size_t vb = (size_t)bi * AJ * 128;
  const int m  = lane & 15;
  const int kb = (lane >> 4) * 2;
  const v8f z = {0.f,0.f,0.f,0.f,0.f,0.f,0.f,0.f};
  v8f acc[8] = {z, z, z, z, z, z, z, z};
  for (int k0 = 0; k0 < AJ; k0 += 4) {
    v2f a = *(const v2f*)(P + pb + (size_t)(j0 + m) * AJ + k0 + kb);
    const float* vr0 = V + vb + (size_t)(k0 + kb) * 128 + m;
#pragma unroll
    for (int t = 0; t < 8; ++t) {
      v2f b;
      b.x = vr0[t * 16];
      b.y = vr0[t * 16 + 128];
      acc[t] = wmma4(a, b, acc[t]);
    }
  }
  const int mo = (lane >> 4) * 8;
#pragma unroll
  for (int t = 0; t < 8; ++t)
#pragma unroll
    for (int r = 0; r < 8; ++r)
      O[vb + (size_t)(j0 + mo + r) * 128 + t * 16 + m] = acc[t][r];
}

// ---------------------------------------------------------------------------
// Unpack h: x_h[b,c,h,w] = O[b,h+1,w,c]+O[b,h,192+w,c];
//           y_h2        = O[b,h+1,96+w,c]+O[b,h,288+w,c]
// ---------------------------------------------------------------------------
__global__ void k_unpack_h(const float* __restrict__ O, float* __restrict__ xo,
                           float* __restrict__ yo) {
  size_t idx = (size_t)blockIdx.x * blockDim.x + threadIdx.x;
  if (idx >= NPIX) return;
  int w = (int)(idx % WW);
  int h = (int)((idx / WW) % HH);
  int c = (int)((idx / HW) % CCH);
  int b = (int)(idx / ((size_t)CCH * HW));
  size_t b97 = (size_t)b * AI;
  xo[idx] = O[((b97 + h + 1) * AJ + w)       * 128 + c] +
            O[((b97 + h)     * AJ + 192 + w) * 128 + c];
  yo[idx] = O[((b97 + h + 1) * AJ + 96 + w)  * 128 + c] +
            O[((b97 + h)     * AJ + 288 + w) * 128 + c];
}

// Unpack v: x_v[b,c,h,w] = O[b,w+1,h,c]+O[b,w,192+h,c];
//           y_v2        = O[b,w+1,96+h,c]+O[b,w,288+h,c]
__global__ void k_unpack_v(const float* __restrict__ O, float* __restrict__ xo,
                           float* __restrict__ yo) {
  size_t idx = (size_t)blockIdx.x * blockDim.x + threadIdx.x;
  if (idx >= NPIX) return;
  int w = (int)(idx % WW);
  int h = (int)((idx / WW) % HH);
  int c = (int)((idx / HW) % CCH);
  int b = (int)(idx / ((size_t)CCH * HW));
  size_t b97 = (size_t)b * AI;
  xo[idx] = O[((b97 + w + 1) * AJ + h)       * 128 + c] +
            O[((b97 + w)     * AJ + 192 + h) * 128 + c];
  yo[idx] = O[((b97 + w + 1) * AJ + 96 + h)  * 128 + c] +
            O[((b97 + w)     * AJ + 288 + h) * 128 + c];
}

// ---------------------------------------------------------------------------
extern "C" void kernel_launch(void* const* d_in, const int* in_sizes, int n_in,
                              void* d_out, int out_size, void* d_ws, size_t ws_size,
                              hipStream_t stream) {
  (void)in_sizes; (void)n_in; (void)out_size; (void)ws_size;
  const float* x = (const float*)d_in[0];
  const float* y = (const float*)d_in[1];
  // conv params: xh,xv,yh,yv,xo,yo each {w,g,b,rm,rv} starting at index 2
  const float* cw[6], *cg[6], *cb[6], *crm[6], *crv[6];
  for (int k = 0; k < 6; ++k) {
    cw[k]  = (const float*)d_in[2 + 5*k + 0];
    cg[k]  = (const float*)d_in[2 + 5*k + 1];
    cb[k]  = (const float*)d_in[2 + 5*k + 2];
    crm[k] = (const float*)d_in[2 + 5*k + 3];
    crv[k] = (const float*)d_in[2 + 5*k + 4];
  }
  // attn params: h then v, each {lng,lnb,wq,wk,wv} starting at index 32
  const float* alng[2], *alnb[2], *awq[2], *awk[2], *awv[2];
  for (int k = 0; k < 2; ++k) {
    alng[k] = (const float*)d_in[32 + 5*k + 0];
    alnb[k] = (const float*)d_in[32 + 5*k + 1];
    awq[k]  = (const float*)d_in[32 + 5*k + 2];
    awk[k]  = (const float*)d_in[32 + 5*k + 3];
    awv[k]  = (const float*)d_in[32 + 5*k + 4];
  }
  float* ws = (float*)d_ws;
  float* C0 = ws;              // xh -> x_h
  float* C1 = C0 + NPIX;       // yh -> y_h2
  float* C2 = C1 + NPIX;       // xv -> x_v
  float* C3 = C2 + NPIX;       // yv -> y_v2
  float* T  = C3 + NPIX;       // attn-layout tensor / xn / attn output
  float* Q  = T + TSZ;
  float* K  = Q + TSZ;
  float* V  = K + TSZ;
  float* S  = V + TSZ;         // scores
  float* outx = (float*)d_out;
  float* outy = outx + NPIX;

  dim3 gConv(HW / 256, CCH / 16, BB), bConv(128);
  dim3 gPack((unsigned)((TSZ + 255) / 256)), bPack(256);
  dim3 gLN((unsigned)(AROWS / 4)), bLN(128);
  dim3 gQKV((unsigned)(AROWS / 16), 2), bQKV(128);
  dim3 gSc(AJ / 64, AJ / 64, BB * AI), bSc(128);
  dim3 gSm((unsigned)(((size_t)BB * AJ * AJ + 255) / 256)), bSm(256);
  dim3 gAV(AJ / 64, BB * AI), bAV(128);
  dim3 gUp((unsigned)((NPIX + 255) / 256)), bUp(256);

  // input convs
  k_conv_bn_gelu_t<false,false><<<gConv, bConv, 0, stream>>>(x, nullptr, cw[0], cg[0], cb[0], crm[0], crv[0], nullptr, C0); // xh
  k_conv_bn_gelu_t<false,false><<<gConv, bConv, 0, stream>>>(x, nullptr, cw[1], cg[1], cb[1], crm[1], crv[1], nullptr, C2); // xv
  k_conv_bn_gelu_t<false,false><<<gConv, bConv, 0, stream>>>(y, nullptr, cw[2], cg[2], cb[2], crm[2], crv[2], nullptr, C1); // yh
  k_conv_bn_gelu_t<false,false><<<gConv, bConv, 0, stream>>>(y, nullptr, cw[3], cg[3], cb[3], crm[3], crv[3], nullptr, C3); // yv

  // ---- h path ----
  k_pack_h<<<gPack, bPack, 0, stream>>>(C0, C1, T);
  k_layernorm<<<gLN, bLN, 0, stream>>>(T, alng[0], alnb[0]);
  k_qkv<<<gQKV, bQKV, 0, stream>>>(T, awq[0], awk[0], awv[0], Q, K, V);
  k_scores<<<gSc, bSc, 0, stream>>>(Q, K, S);
  k_softmax_i<<<gSm, bSm, 0, stream>>>(S);
  k_av<<<gAV, bAV, 0, stream>>>(S, V, T);         // T now holds attn output
  k_unpack_h<<<gUp, bUp, 0, stream>>>(T, C0, C1); // C0=x_h, C1=y_h2

  // ---- v path ----
  k_pack_v<<<gPack, bPack, 0, stream>>>(C2, C3, T);
  k_layernorm<<<gLN, bLN, 0, stream>>>(T, alng[1], alnb[1]);
  k_qkv<<<gQKV, bQKV, 0, stream>>>(T, awq[1], awk[1], awv[1], Q, K, V);
  k_scores<<<gSc, bSc, 0, stream>>>(Q, K, S);
  k_softmax_i<<<gSm, bSm, 0, stream>>>(S);
  k_av<<<gAV, bAV, 0, stream>>>(S, V, T);
  k_unpack_v<<<gUp, bUp, 0, stream>>>(T, C2, C3); // C2=x_v, C3=y_v2

  // ---- output convs: gelu(bn(W*(h+v))) + residual ----
  k_conv_bn_gelu_t<true,true><<<gConv, bConv, 0, stream>>>(C0, C2, cw[4], cg[4], cb[4], crm[4], crv[4], x, outx);
  k_conv_bn_gelu_t<true,true><<<gConv, bConv, 0, stream>>>(C1, C3, cw[5], cg[5], cb[5], crm[5], crv[5], y, outy);
}